// l0_MFSIT_Net_39900246180384
// MI455X (gfx1250) — compile-verified
//
#include <hip/hip_runtime.h>
#include <hip/hip_bf16.h>

#define NN   471
#define MM   128
#define TT   16384
#define TOPK 50
#define NPAD 480
#define ROWS_PER_BLK 64
#define NBLK (TT / ROWS_PER_BLK)   // 256
#define KMAIN (NN & ~3)            // 468: last full 4-wide K step boundary

typedef float v2f __attribute__((ext_vector_type(2)));
typedef float v8f __attribute__((ext_vector_type(8)));

// ---------------- Kernel 1: deterministic partial column sums of x ----------------
__global__ __launch_bounds__(256)
void k_colsum(const float* __restrict__ x, float* __restrict__ partial) {
    const int tid = threadIdx.x;                 // 256 threads
    const size_t r0 = (size_t)blockIdx.x * ROWS_PER_BLK;
    float acc0 = 0.f, acc1 = 0.f;
    const int n1 = tid + 256;
    for (int r = 0; r < ROWS_PER_BLK; ++r) {
        const float* row = x + (r0 + r) * NN;
        __builtin_prefetch(row + NN, 0, 1);      // global_prefetch_b8: next row
        acc0 += row[tid];
        if (n1 < NN) acc1 += row[n1];
    }
    partial[(size_t)blockIdx.x * NN + tid] = acc0;
    if (n1 < NN) partial[(size_t)blockIdx.x * NN + n1] = acc1;
}

// ---------------- Kernel 2: finish mean over T ----------------
__global__ __launch_bounds__(512)
void k_meanfin(const float* __restrict__ partial, float* __restrict__ meanx) {
    const int n = threadIdx.x;
    if (n < NN) {
        float s = 0.f;
        for (int b = 0; b < NBLK; ++b) s += partial[(size_t)b * NN + n];
        meanx[n] = s * (1.0f / (float)TT);
    }
}

// ---------------- Kernel 3: persistent single-workgroup solver ----------------
__global__ __launch_bounds__(512)
void k_solver(const float* __restrict__ qt,      // [128,471]
              const float* __restrict__ w,       // [471]
              const float* __restrict__ alpha_p,
              const float* __restrict__ lamda_p,
              const float* __restrict__ rho_p,
              const float* __restrict__ mu_p,
              const float* __restrict__ lw,      // [471,471]
              const float* __restrict__ lb,      // [471]
              const float* __restrict__ meanx,   // [471]
              float* __restrict__ out) {         // [471]
    __shared__ float G[MM * MM];     // 64 KB: G then G^{-1} in place
    __shared__ float prow[MM];
    __shared__ float theta[MM];
    __shared__ float t1[MM];
    __shared__ float mx[NPAD];
    __shared__ float wv[NPAD], zv[NPAD], uv[NPAD];
    __shared__ float W2v[NPAD], bvec[NPAD], znew[NPAD], vv[NPAD];
    __shared__ float red[512];

    const int tid  = threadIdx.x;        // 512 threads = 16 wave32
    const int lane = tid & 31;
    const int wave = tid >> 5;

    const float alpha = alpha_p[0];
    const float lamda = lamda_p[0];
    const float rho   = rho_p[0];
    const float mu    = mu_p[0];

    if (tid < NPAD) {
        mx[tid] = (tid < NN) ? meanx[tid] : 0.f;
        wv[tid] = (tid < NN) ? w[tid]     : 0.f;
        zv[tid] = 0.f; uv[tid] = 0.f; vv[tid] = 0.f;
        W2v[tid] = 0.f; bvec[tid] = 0.f; znew[tid] = 0.f;
    }
    __syncthreads();

    // ---- Phase 1: W2 = linear_w @ mean_x + linear_b (wave-per-row dots) ----
    for (int row = wave; row < NN; row += 16) {
        const float* lr = lw + (size_t)row * NN;
        float a = 0.f;
        for (int k = lane; k < NN; k += 32) a += lr[k] * mx[k];
        for (int off = 16; off > 0; off >>= 1) a += __shfl_down(a, off, 32);
        if (lane == 0) W2v[row] = a + lb[row];
    }

    // ---- Phase 2: G = q_t q_t^T via V_WMMA_F32_16X16X4_F32 ----
    // 8x8 = 64 output tiles of 16x16; 4 tiles per wave.
    // A (16x4 f32): VGPR v, lane l -> M = l&15, K = k + v + 2*(l>>4)
    // B (4x16 f32): VGPR v, lane l -> N = l&15, K = k + v + 2*(l>>4)  (B[k][n] = qt[tj*16+n][k])
    for (int tile = wave; tile < 64; tile += 16) {
        const int ti = tile >> 3, tj = tile & 7;
        const float* arow = qt + (size_t)(ti * 16 + (lane & 15)) * NN;
        const float* brow = qt + (size_t)(tj * 16 + (lane & 15)) * NN;
        const int kb = 2 * (lane >> 4);
        v8f c = {};
        // Main K loop: fully in-bounds, no divergence -> clean b64 loads + wmma
        for (int k = 0; k < KMAIN; k += 4) {
            const int k0 = k + kb;
            v2f a, b;
            a.x = arow[k0];
            a.y = arow[k0 + 1];
            b.x = brow[k0];
            b.y = brow[k0 + 1];
            c = __builtin_amdgcn_wmma_f32_16x16x4_f32(false, a, false, b,
                                                      (short)0, c, false, false);
        }
        // Peeled tail: K = 468..470 (+1 zero pad lane-wise)
        {
            const int k0 = KMAIN + kb;           // 468 or 470, always < NN
            v2f a, b;
            a.x = arow[k0];
            b.x = brow[k0];
            const bool in = (k0 + 1) < NN;       // only lanes with kb==0
            a.y = in ? arow[k0 + 1] : 0.f;
            b.y = in ? brow[k0 + 1] : 0.f;
            c = __builtin_amdgcn_wmma_f32_16x16x4_f32(false, a, false, b,
                                                      (short)0, c, false, false);
        }
        // C/D layout: VGPR r, lane l -> row = r + 8*(l>>4), col = l&15
        const int rbase = ti * 16 + 8 * (lane >> 4);
        const int col   = tj * 16 + (lane & 15);
        #pragma unroll
        for (int r = 0; r < 8; ++r) G[(rbase + r) * MM + col] = c[r];
    }
    __syncthreads();

    // ---- Phase 3: in-place Gauss-Jordan inversion of G (SPD -> no pivoting) ----
    const int gr  = tid >> 2;            // row 0..127 (4 threads per row)
    const int gc0 = (tid & 3) * 32;      // 32-column chunk
    for (int p = 0; p < MM; ++p) {
        const float f = G[gr * MM + p];          // read before any writes this step
        if (tid < MM) prow[tid] = G[p * MM + tid];
        __syncthreads();
        const float invp = 1.0f / prow[p];
        if (gr == p) {
            for (int j = gc0; j < gc0 + 32; ++j)
                G[p * MM + j] = (j == p) ? invp : prow[j] * invp;
        } else {
            for (int j = gc0; j < gc0 + 32; ++j)
                G[gr * MM + j] = (j == p) ? (-f * invp)
                                          : (G[gr * MM + j] - f * (prow[j] * invp));
        }
        __syncthreads();
    }

    // ---- Phase 4: 10 ADMM iterations ----
    for (int it = 0; it < 10; ++it) {
        // v = u - rho*(z - w)
        if (tid < NN) vv[tid] = uv[tid] - rho * (zv[tid] - wv[tid]);
        __syncthreads();

        // t1 = q_t @ v   [128]
        for (int row = wave; row < MM; row += 16) {
            const float* qr = qt + (size_t)row * NN;
            float a = 0.f;
            for (int k = lane; k < NN; k += 32) a += qr[k] * vv[k];
            for (int off = 16; off > 0; off >>= 1) a += __shfl_down(a, off, 32);
            if (lane == 0) t1[row] = a;
        }
        __syncthreads();

        // theta = alpha * (G^{-1} @ t1)   [128]
        for (int row = wave; row < MM; row += 16) {
            float a = 0.f;
            for (int k = lane; k < MM; k += 32) a += G[row * MM + k] * t1[k];
            for (int off = 16; off > 0; off >>= 1) a += __shfl_down(a, off, 32);
            if (lane == 0) theta[row] = alpha * a;
        }
        __syncthreads();

        // b = w + (q_t^T @ theta)/N   (coalesced column reads)
        if (tid < NN) {
            float a = 0.f;
            for (int m2 = 0; m2 < MM; ++m2) a += qt[(size_t)m2 * NN + tid] * theta[m2];
            bvec[tid] = wv[tid] + a * (1.0f / (float)NN);
        }
        __syncthreads();

        // sum(z) (old z)
        red[tid] = (tid < NN) ? zv[tid] : 0.f;
        for (int s = 256; s > 0; s >>= 1) {
            __syncthreads();
            if (tid < s) red[tid] += red[tid + s];
        }
        __syncthreads();
        const float sz = red[0];

        // grad + relu
        if (tid < NN) {
            const float z = zv[tid];
            float g = W2v[tid] + rho * (z - bvec[tid]) + uv[tid]
                    + 2.0f * lamda * (sz - 1.0f)
                    + 2.0f * lamda * fminf(0.0f, z);
            znew[tid] = fmaxf(0.0f, z - mu * g);
        } else {
            znew[tid] = 0.f;
        }
        __syncthreads();

        // exact top-k mask via rank counting (ties -> lower index, as lax.top_k)
        if (tid < NN) {
            const float zi = znew[tid];
            int rank = 0;
            for (int j = 0; j < NN; ++j) {
                const float zj = znew[j];
                rank += (zj > zi) || ((zj == zi) && (j < tid));
            }
            zv[tid] = (rank < TOPK) ? zi : 0.f;
        }
        __syncthreads();

        // u += rho*(z_new - b)  (skipped on the final extra (theta,z) step)
        if (it < 9) {
            if (tid < NN) uv[tid] += rho * (zv[tid] - bvec[tid]);
            __syncthreads();
        }
    }

    // ---- normalize: out = z / (sum(z) + 1e-8) ----
    red[tid] = (tid < NN) ? zv[tid] : 0.f;
    for (int s = 256; s > 0; s >>= 1) {
        __syncthreads();
        if (tid < s) red[tid] += red[tid + s];
    }
    __syncthreads();
    const float S = red[0];
    if (tid < NN) out[tid] = zv[tid] / (S + 1e-8f);
}

// ---------------- Host launcher ----------------
extern "C" void kernel_launch(void* const* d_in, const int* in_sizes, int n_in,
                              void* d_out, int out_size, void* d_ws, size_t ws_size,
                              hipStream_t stream) {
    const float* x     = (const float*)d_in[0];   // [16384,471]
    const float* qt    = (const float*)d_in[1];   // [128,471]
    const float* w     = (const float*)d_in[2];   // [471]
    // d_in[3] = b1 (unused by forward math)
    const float* alpha = (const float*)d_in[4];
    const float* lamda = (const float*)d_in[5];
    const float* rho   = (const float*)d_in[6];
    const float* mu    = (const float*)d_in[7];
    const float* lw    = (const float*)d_in[8];   // [471,471]
    const float* lb    = (const float*)d_in[9];   // [471]
    float* out = (float*)d_out;

    float* ws      = (float*)d_ws;
    float* partial = ws;                          // 256*471 floats
    float* meanx   = ws + (size_t)NBLK * NN;      // 471 floats

    k_colsum <<<NBLK, 256, 0, stream>>>(x, partial);
    k_meanfin<<<1,    512, 0, stream>>>(partial, meanx);
    k_solver <<<1,    512, 0, stream>>>(qt, w, alpha, lamda, rho, mu, lw, lb, meanx, out);
}